// CTRModel_47339129537131
// MI455X (gfx1250) — compile-verified
//
#include <hip/hip_runtime.h>
#include <hip/hip_bf16.h>

#define N_NODE   100000
#define N_RELS   64
#define DIMK     64
#define N_FACTOR 4
#define N_LAYERS 2
#define BATCH    2048
#define NEIGH    32
#define ROWS     (NEIGH * N_FACTOR)   /* 128 rows = K*F per batch element */

typedef __attribute__((ext_vector_type(2))) float v2f;
typedef __attribute__((ext_vector_type(8))) float v8f;

// ---------------------------------------------------------------------------
// Precompute R_pre[rel][f][d] = sum_e relation_emb[rel,f,e] * W1[64+e, d] + b1[d]
// Only 64 relations exist -> folds the r-half of the concat GEMM + bias into a
// 64KB table, halving the per-batch WMMA work.
// ---------------------------------------------------------------------------
__global__ __launch_bounds__(256)
void ctr_rpre_kernel(const float* __restrict__ rel_emb,
                     const float* __restrict__ W1,
                     const float* __restrict__ b1,
                     float* __restrict__ rpre) {
    int g  = blockIdx.x * 256 + threadIdx.x;   // 0 .. 16383
    int d  = g & 63;
    int fe = g >> 6;                           // rel*4 + f
    const float* x = rel_emb + fe * 64;
    float acc = b1[d];
#pragma unroll 8
    for (int e = 0; e < 64; ++e)
        acc = fmaf(x[e], W1[(64 + e) * 64 + d], acc);
    rpre[g] = acc;
}

// ---------------------------------------------------------------------------
// Layer-0 output: out[side, 0, b, :, :] = node_emb[base_idx[b]]
// ---------------------------------------------------------------------------
__global__ __launch_bounds__(256)
void ctr_layer0_kernel(const int* __restrict__ users,
                       const int* __restrict__ items,
                       const float* __restrict__ node_emb,
                       float* __restrict__ out) {
    int b    = blockIdx.x;
    int side = blockIdx.y;
    const int* idx = side ? items : users;
    size_t src = (size_t)idx[b] * 256 + threadIdx.x;
    out[(size_t)side * 3 * BATCH * 256 + (size_t)b * 256 + threadIdx.x] = node_emb[src];
}

// ---------------------------------------------------------------------------
// Main attention kernel: one 256-thread block (8 wave32) per (b, layer, side).
//   hid = relu(h @ W1h + R_pre[r])        via V_WMMA_F32_16X16X4_F32
//   att = hid . W2 + b2                   via shuffle reduction
//   w   = softmax_k(att)                  in LDS
//   out = sum_k w[k,f] * node_emb[t_k,f]  coalesced gather
//
// W1h is staged in LDS *pre-swizzled into B-fragment order*: for k-step kk,
// n-tile t, lane l the pair (W1h[ka][col], W1h[ka+1][col]) with
// ka = kk*4 + (l>>4)*2, col = t*16 + (l&15) lives at one contiguous v2f, so
// the GEMM inner loop issues a single conflict-free ds_load_b64 per fragment
// and WMMA sources land in even-aligned VGPR pairs with no repack moves.
// ---------------------------------------------------------------------------
__global__ __launch_bounds__(256)
void ctr_attn_kernel(const int* __restrict__ users_h, const int* __restrict__ users_r,
                     const int* __restrict__ users_t,
                     const int* __restrict__ items_h, const int* __restrict__ items_r,
                     const int* __restrict__ items_t,
                     const float* __restrict__ node_emb,
                     const float* __restrict__ W1,
                     const float* __restrict__ W2,
                     const float* __restrict__ b2,
                     const float* __restrict__ rpre,
                     float* __restrict__ out) {
    __shared__ float s_w1b[16 * 4 * 64];  // 16 KB : W1h in B-fragment order
    __shared__ float s_h[ROWS * 64];      // 32 KB : gathered h rows, m = k*4+f
    __shared__ float s_att[ROWS];
    __shared__ float s_w[ROWS];
    __shared__ float s_mx[N_FACTOR];
    __shared__ float s_sm[N_FACTOR];
    __shared__ int   s_tidx[NEIGH];

    const int tid   = threadIdx.x;
    const int b     = blockIdx.x;
    const int layer = blockIdx.y;
    const int side  = blockIdx.z;

    const int* hI = side ? items_h : users_h;
    const int* rI = side ? items_r : users_r;
    const int* tI = side ? items_t : users_t;
    const size_t ibase = (size_t)layer * BATCH * NEIGH + (size_t)b * NEIGH;

    // ---- stage W1h swizzled into B-fragment order (16 floats per thread)
    // flat g = ((kk*4 + t)*32 + lane)*2 + c  <-  W1[e*64 + d],
    //   e = kk*4 + (lane>>4)*2 + c,  d = t*16 + (lane&15)
    for (int g = tid; g < 4096; g += 256) {
        int c    = g & 1;
        int ln   = (g >> 1) & 31;
        int t    = (g >> 6) & 3;
        int kk   = g >> 8;
        int e    = kk * 4 + ((ln >> 4) << 1) + c;
        int d    = t * 16 + (ln & 15);
        s_w1b[g] = W1[e * 64 + d];
    }

    // ---- stage the 128 gathered h-rows: 2 threads per row, 8x float4 each
    {
        int row   = tid >> 1;
        int halfR = tid & 1;
        int k = row >> 2, f = row & 3;
        const float* src = node_emb + (size_t)hI[ibase + k] * 256 + f * 64 + halfR * 32;
        float*       dst = s_h + row * 64 + halfR * 32;
#pragma unroll
        for (int q = 0; q < 8; ++q)
            ((float4*)dst)[q] = ((const float4*)src)[q];
    }
    if (tid < NEIGH) s_tidx[tid] = tI[ibase + tid];
    __syncthreads();

    // ---- 128x64x64 GEMM with V_WMMA_F32_16X16X4_F32
    const int wave = tid >> 5;
    const int lane = tid & 31;
    const int half = lane >> 4;     // 0: lanes 0-15 (K pair 0,1)  1: lanes 16-31 (K pair 2,3)
    const int l15  = lane & 15;
    const int arow = wave * 16 + l15;

    float w2l[4];
#pragma unroll
    for (int t = 0; t < 4; ++t) w2l[t] = W2[t * 16 + l15];
    const float b2v = b2[0];

    v8f z = {};
    v8f acc[4] = {z, z, z, z};

#pragma unroll 4
    for (int kk = 0; kk < 16; ++kk) {
        const int ka = kk * 4 + half * 2;
        const v2f a = *(const v2f*)&s_h[arow * 64 + ka];           // one ds_load_b64
#pragma unroll
        for (int t = 0; t < 4; ++t) {
            const v2f bf = *(const v2f*)&s_w1b[((kk * 4 + t) * 32 + lane) * 2]; // one ds_load_b64
            acc[t] = __builtin_amdgcn_wmma_f32_16x16x4_f32(
                /*neg_a=*/false, a, /*neg_b=*/false, bf,
                /*c_mod=*/(short)0, acc[t], /*reuse_a=*/false, /*reuse_b=*/false);
        }
    }

    // ---- hid = relu(gemm + R_pre[r_idx]); att = hid . W2 + b2
    // D layout: acc[t] vgpr j, lanes 0-15 -> row wave*16+j, lanes 16-31 -> row wave*16+8+j
    const int k0    = wave * 4 + half * 2;           // neighbor k for j=0..3
    const int ridx0 = rI[ibase + k0];
    const int ridx1 = rI[ibase + k0 + 1];            // neighbor k for j=4..7

#pragma unroll
    for (int j = 0; j < 8; ++j) {
        const int row  = wave * 16 + half * 8 + j;
        const int f    = j & 3;                      // == row & 3
        const int ridx = (j < 4) ? ridx0 : ridx1;
        const float* rrow = rpre + ridx * 256 + f * 64;
        float part = 0.0f;
#pragma unroll
        for (int t = 0; t < 4; ++t) {
            float hid = acc[t][j] + rrow[t * 16 + l15];
            hid = fmaxf(hid, 0.0f);
            part = fmaf(hid, w2l[t], part);
        }
        // reduce across the 16 lanes of this half (xor masks < 16 stay in-half)
#pragma unroll
        for (int off = 1; off < 16; off <<= 1)
            part += __shfl_xor(part, off, 32);
        if (l15 == 0) s_att[row] = part + b2v;
    }
    __syncthreads();

    // ---- softmax over neighbors per factor
    if (tid < N_FACTOR) {
        float m = -1e30f;
        for (int k = 0; k < NEIGH; ++k) m = fmaxf(m, s_att[k * 4 + tid]);
        s_mx[tid] = m;
        float s = 0.0f;
        for (int k = 0; k < NEIGH; ++k) s += expf(s_att[k * 4 + tid] - m);
        s_sm[tid] = s;
    }
    __syncthreads();
    if (tid < ROWS) {
        int f = tid & 3;
        s_w[tid] = expf(s_att[tid] - s_mx[f]) / s_sm[f];
    }
    __syncthreads();

    // ---- weighted sum of t-embeddings; per-k read is a contiguous 1KB row
    {
        const int f = tid >> 6;
        const int d = tid & 63;
        float accv = 0.0f;
#pragma unroll 4
        for (int k = 0; k < NEIGH; ++k)
            accv = fmaf(s_w[k * 4 + f],
                        node_emb[(size_t)s_tidx[k] * 256 + f * 64 + d], accv);
        size_t ob = (size_t)side * 3 * BATCH * 256
                  + (size_t)(layer + 1) * BATCH * 256
                  + (size_t)b * 256 + tid;
        out[ob] = accv;
    }
}

// ---------------------------------------------------------------------------
extern "C" void kernel_launch(void* const* d_in, const int* in_sizes, int n_in,
                              void* d_out, int out_size, void* d_ws, size_t ws_size,
                              hipStream_t stream) {
    (void)in_sizes; (void)n_in; (void)out_size; (void)ws_size;

    const int*   users    = (const int*)  d_in[0];
    const int*   items    = (const int*)  d_in[1];
    const int*   users_h  = (const int*)  d_in[2];
    const int*   users_r  = (const int*)  d_in[3];
    const int*   users_t  = (const int*)  d_in[4];
    const int*   items_h  = (const int*)  d_in[5];
    const int*   items_r  = (const int*)  d_in[6];
    const int*   items_t  = (const int*)  d_in[7];
    const float* node_emb = (const float*)d_in[8];
    const float* rel_emb  = (const float*)d_in[9];
    const float* W1       = (const float*)d_in[10];
    const float* b1       = (const float*)d_in[11];
    const float* W2       = (const float*)d_in[12];
    const float* b2       = (const float*)d_in[13];

    float* out  = (float*)d_out;
    float* rpre = (float*)d_ws;          // 64*4*64 floats = 64 KB scratch

    ctr_rpre_kernel  <<<64, 256, 0, stream>>>(rel_emb, W1, b1, rpre);
    ctr_layer0_kernel<<<dim3(BATCH, 2), 256, 0, stream>>>(users, items, node_emb, out);
    ctr_attn_kernel  <<<dim3(BATCH, N_LAYERS, 2), 256, 0, stream>>>(
        users_h, users_r, users_t, items_h, items_r, items_t,
        node_emb, W1, W2, b2, rpre, out);
}